// LogSigLayer_42545946034347
// MI455X (gfx1250) — compile-verified
//
#include <hip/hip_runtime.h>

typedef __attribute__((ext_vector_type(2))) float v2f;
typedef __attribute__((ext_vector_type(8))) float v8f;

#define LROWS 2048
#define DIMS  32
#define KTOT  2047   // L-1 inner-product terms
#define OUTD  528    // 32 level1 + 496 areas
#define WAVES 4
#define KCHUNK 512   // K slice per wave (ceil(2047/4) rounded to mult of 4)

__global__ __launch_bounds__(128) void logsig2_wmma_kernel(
    const float* __restrict__ x, float* __restrict__ out) {
  const int b = blockIdx.x;
  const float* xbase = x + (size_t)b * LROWS * DIMS;

  const int tid  = threadIdx.x;
  const int lane = tid & 31;
  const int wave = tid >> 5;
  const int half = lane >> 4;   // K-group within fragment (0: k0,k0+1; 1: k0+2,k0+3)
  const int t    = lane & 15;   // M (for A frags) / N (for B frags) index within tile

  // Basepoint x[0] at columns t and t+16 (prefix_excl = x[k] - x[0])
  const float x0a = xbase[t];
  const float x0b = xbase[t + 16];

  v8f acc00 = {}; v8f acc01 = {}; v8f acc10 = {}; v8f acc11 = {};

  const int kbase = wave * KCHUNK;
  const int kend  = (kbase + KCHUNK < KTOT) ? (kbase + KCHUNK) : KTOT;

  for (int k0 = kbase; k0 < kend; k0 += 4) {
    const int r0 = k0 + 2 * half;
    const int r1 = r0 + 1;
    // Clamp: at the global tail (r0==2046) this makes dx of slot k=2047 exactly 0,
    // zeroing its WMMA contribution without breaking EXEC=all-ones.
    const int r2 = (r0 + 2 < LROWS) ? (r0 + 2) : (LROWS - 1);

    const float a0 = xbase[r0 * DIMS + t];
    const float a1 = xbase[r1 * DIMS + t];
    const float a2 = xbase[r2 * DIMS + t];
    const float b0 = xbase[r0 * DIMS + t + 16];
    const float b1 = xbase[r1 * DIMS + t + 16];
    const float b2 = xbase[r2 * DIMS + t + 16];

    // Stream next step's rows toward the caches.
    __builtin_prefetch(&xbase[(r0 + 4) * DIMS + t], 0, 0);

    v2f Alo, Ahi, Blo, Bhi;
    Alo[0] = a0 - x0a;  Alo[1] = a1 - x0a;   // prefix_excl, M tile 0..15
    Ahi[0] = b0 - x0b;  Ahi[1] = b1 - x0b;   // prefix_excl, M tile 16..31
    Blo[0] = a1 - a0;   Blo[1] = a2 - a1;    // dx, N tile 0..15
    Bhi[0] = b1 - b0;   Bhi[1] = b2 - b1;    // dx, N tile 16..31

    acc00 = __builtin_amdgcn_wmma_f32_16x16x4_f32(false, Alo, false, Blo,
                                                  (short)0, acc00, false, false);
    acc01 = __builtin_amdgcn_wmma_f32_16x16x4_f32(false, Alo, false, Bhi,
                                                  (short)0, acc01, false, false);
    acc10 = __builtin_amdgcn_wmma_f32_16x16x4_f32(false, Ahi, false, Blo,
                                                  (short)0, acc10, false, false);
    acc11 = __builtin_amdgcn_wmma_f32_16x16x4_f32(false, Ahi, false, Bhi,
                                                  (short)0, acc11, false, false);
  }

  // Spill per-wave partial S2 (32x32) to LDS. C/D layout: VGPR r holds
  // M = r + 8*half, N = t (within a 16x16 tile).
  __shared__ float s2[WAVES][DIMS][DIMS];   // 16 KiB
#pragma unroll
  for (int r = 0; r < 8; ++r) {
    const int mrow = r + 8 * half;
    s2[wave][mrow +  0][t +  0] = acc00[r];
    s2[wave][mrow +  0][t + 16] = acc01[r];
    s2[wave][mrow + 16][t +  0] = acc10[r];
    s2[wave][mrow + 16][t + 16] = acc11[r];
  }
  __syncthreads();

  float* ob = out + (size_t)b * OUTD;

  // level1 = x[L-1] - x[0]
  if (tid < DIMS) {
    ob[tid] = xbase[(LROWS - 1) * DIMS + tid] - xbase[tid];
  }

  // areas = 0.5*(S2 - S2^T) at triu indices (k=1), row-major pair order
  for (int p = tid; p < 496; p += 128) {
    int i = 0, rem = p;
    while (rem >= (DIMS - 1 - i)) { rem -= (DIMS - 1 - i); ++i; }
    const int j = i + 1 + rem;
    float sij = 0.f, sji = 0.f;
#pragma unroll
    for (int w = 0; w < WAVES; ++w) {
      sij += s2[w][i][j];
      sji += s2[w][j][i];
    }
    ob[DIMS + p] = 0.5f * (sij - sji);
  }
}

extern "C" void kernel_launch(void* const* d_in, const int* in_sizes, int n_in,
                              void* d_out, int out_size, void* d_ws, size_t ws_size,
                              hipStream_t stream) {
  const float* x = (const float*)d_in[0];
  float* out = (float*)d_out;
  const int B = in_sizes[0] / (LROWS * DIMS);   // 512
  logsig2_wmma_kernel<<<dim3(B), dim3(128), 0, stream>>>(x, out);
}